// SparseMoE_69234872811961
// MI455X (gfx1250) — compile-verified
//
#include <hip/hip_runtime.h>
#include <hip/hip_bf16.h>
#include <math.h>

// Problem constants (match reference): H=2048, I=8192, E=8, top_k=2, T=B*S=4096
constexpr int kH = 2048;
constexpr int kI = 8192;
constexpr int kE = 8;
constexpr int kT = 4096;

constexpr int MT = 64;    // tokens per FFN tile (4 x 16 row-subtiles)
constexpr int IC = 128;   // I-dimension chunk staged as hmid in LDS
constexpr int KC = 64;    // H-dimension chunk of X staged in LDS
constexpr int NB = 512;   // out columns per block (H / 4 column groups)

typedef __attribute__((ext_vector_type(2))) float v2f;
typedef __attribute__((ext_vector_type(8))) float v8f;

__device__ __forceinline__ v8f wmma_f32_16x16x4(v2f a, v2f b, v8f c) {
  // (neg_a, A, neg_b, B, c_mod, C, reuse_a, reuse_b)
  return __builtin_amdgcn_wmma_f32_16x16x4_f32(false, a, false, b, (short)0, c,
                                               false, false);
}

__device__ __forceinline__ float gelu_exact(float v) {
  return 0.5f * v * (1.0f + erff(v * 0.70710678118654752f));
}

// out[t*H + h] = bias[h]  (accumulation base; bias add from reference)
__global__ void moe_init_out(float* __restrict__ out,
                             const float* __restrict__ bias) {
  size_t i = (size_t)blockIdx.x * blockDim.x + threadIdx.x;
  if (i < (size_t)kT * kH) out[i] = bias[i & (kH - 1)];
}

// One thread per token: router logits, softmax, top-2 (first-index tie-break),
// normalized combine weights.
__global__ void moe_router(const float* __restrict__ x,
                           const float* __restrict__ gate_w,
                           float* __restrict__ logits_out,
                           float* __restrict__ combine) {
  int t = blockIdx.x * blockDim.x + threadIdx.x;
  if (t >= kT) return;
  float acc[kE];
#pragma unroll
  for (int e = 0; e < kE; ++e) acc[e] = 0.0f;
  const float* xr = x + (size_t)t * kH;
  for (int h = 0; h < kH; ++h) {
    float xv = xr[h];
    const float* gr = gate_w + (size_t)h * kE;
#pragma unroll
    for (int e = 0; e < kE; ++e) acc[e] += xv * gr[e];
  }
#pragma unroll
  for (int e = 0; e < kE; ++e) logits_out[(size_t)t * kE + e] = acc[e];
  float m = acc[0];
#pragma unroll
  for (int e = 1; e < kE; ++e) m = fmaxf(m, acc[e]);
  float p[kE], s = 0.0f;
#pragma unroll
  for (int e = 0; e < kE; ++e) { p[e] = expf(acc[e] - m); s += p[e]; }
  float inv = 1.0f / s;
#pragma unroll
  for (int e = 0; e < kE; ++e) p[e] *= inv;
  int i0 = 0; float v0 = p[0];
#pragma unroll
  for (int e = 1; e < kE; ++e) if (p[e] > v0) { v0 = p[e]; i0 = e; }
  int i1 = -1; float v1 = -1.0f;
#pragma unroll
  for (int e = 0; e < kE; ++e)
    if (e != i0 && p[e] > v1) { v1 = p[e]; i1 = e; }
  float rs = 1.0f / (v0 + v1);
#pragma unroll
  for (int e = 0; e < kE; ++e) combine[(size_t)t * kE + e] = 0.0f;
  combine[(size_t)t * kE + i0] = v0 * rs;
  combine[(size_t)t * kE + i1] = v1 * rs;
}

// Deterministic, order-preserving token compaction: one wave32 per expert.
// idx[e*T + 0..cnt) = token ids routed to e (ascending), padded to MT with -1.
__global__ void moe_gather(const float* __restrict__ combine,
                           int* __restrict__ cnts,
                           int* __restrict__ idx) {
  const int e = threadIdx.x >> 5;      // 8 waves, one per expert
  const int lane = threadIdx.x & 31;
  if (e >= kE) return;
  int count = 0;
  const unsigned lt = (1u << lane) - 1u;
  for (int t0 = 0; t0 < kT; t0 += 32) {
    int t = t0 + lane;
    bool sel = combine[(size_t)t * kE + e] != 0.0f;
    unsigned mask = __builtin_amdgcn_ballot_w32(sel);
    if (sel) {
      int pos = count + __builtin_popcount(mask & lt);
      idx[e * kT + pos] = t;
    }
    count += __builtin_popcount(mask);
  }
  int padded = (count + MT - 1) & ~(MT - 1);
  for (int p = count + lane; p < padded; p += 32) idx[e * kT + p] = -1;
  if (lane == 0) cnts[e] = count;
}

// grid = (T/MT worst-case tiles, H/NB column groups, E experts)
// block = 512 threads = 16 wave32.  Fused FFN over gathered tokens with
// V_WMMA_F32_16X16X4_F32; X and hmid staged in LDS.
__global__ __launch_bounds__(512)
void moe_expert_ffn(const float* __restrict__ x,
                    const float* __restrict__ w_in,
                    const float* __restrict__ w_out,
                    const float* __restrict__ combine,
                    const int* __restrict__ cnts,
                    const int* __restrict__ idx,
                    float* __restrict__ out) {
  __shared__ float xs[MT * KC];     // 16 KB staged X chunk
  __shared__ float hmid[MT * IC];   // 32 KB staged gelu(X@W_in) chunk
  __shared__ int   toks[MT];        // clamped gathered token ids
  __shared__ float cmbs[MT];        // per-row combine scale (0 for padding)

  const int e    = blockIdx.z;
  const int cg   = blockIdx.y;          // column group: cols [cg*NB, cg*NB+NB)
  const int tile = blockIdx.x;

  const int cnt = cnts[e];
  if (tile * MT >= cnt) return;         // block-uniform sparsity skip

  if (threadIdx.x < MT) {
    int raw = idx[e * kT + tile * MT + threadIdx.x];
    toks[threadIdx.x] = raw < 0 ? 0 : raw;
    cmbs[threadIdx.x] = raw < 0 ? 0.0f : combine[(size_t)raw * kE + e];
  }
  __syncthreads();

  const int lane = threadIdx.x & 31;
  const int wv   = threadIdx.x >> 5;    // wave 0..15
  const int half = lane >> 4;           // K-half select (ISA A/B layout)
  const int ln   = lane & 15;
  const int cA   = wv & 7;              // GEMM1: wave's column tile in I-chunk
  const int rp   = wv >> 3;             // GEMM1: wave's row pair (0/1)
  const int nb   = cg * NB;

  const float* wi = w_in  + (size_t)e * kH * kI;   // [H, I] row-major
  const float* wo = w_out + (size_t)e * kI * kH;   // [I, H] row-major

  v8f acc[4][2];                        // persistent out C-file: 64 VGPRs
#pragma unroll
  for (int rt = 0; rt < 4; ++rt)
#pragma unroll
    for (int ct = 0; ct < 2; ++ct) acc[rt][ct] = (v8f){};

  const int srow = threadIdx.x >> 3;              // staging: row, 8 cols/thread
  const int scol = (threadIdx.x & 7) * 8;
  const float* srcrow = x + (size_t)toks[srow] * kH + scol;

  for (int c0 = 0; c0 < kI; c0 += IC) {
    // prefetch next chunk's w_out rows (global_prefetch_b8)
    if (c0 + IC < kI)
      __builtin_prefetch(wo + (size_t)(c0 + IC) * kH + nb + threadIdx.x, 0, 3);

    // ---- GEMM1: hmid[MT, IC] = gelu(Xg[MT, H] @ w_in[H, c0:c0+IC])
    v8f c1a = (v8f){};
    v8f c1b = (v8f){};
    const int colI = c0 + cA * 16 + ln;
    for (int ks = 0; ks < kH; ks += KC) {
      {   // cooperatively stage X[MT, KC] (b128 in / b128 to LDS)
        const float* src = srcrow + ks;
        float4 p0 = *(const float4*)src;
        float4 p1 = *(const float4*)(src + 4);
        *(float4*)&xs[srow * KC + scol]     = p0;
        *(float4*)&xs[srow * KC + scol + 4] = p1;
      }
      __syncthreads();
      for (int kb = 0; kb < KC; kb += 4) {
        const int kk = kb + 2 * half;
        const int kg = ks + kk;
        v2f b; b.x = wi[(size_t)kg * kI + colI];
               b.y = wi[(size_t)(kg + 1) * kI + colI];
        v2f a0 = *(const v2f*)&xs[(32 * rp + ln) * KC + kk];
        v2f a1 = *(const v2f*)&xs[(32 * rp + 16 + ln) * KC + kk];
        c1a = wmma_f32_16x16x4(a0, b, c1a);
        c1b = wmma_f32_16x16x4(a1, b, c1b);
      }
      __syncthreads();
    }
    // exact GELU + stage to LDS (C layout: vgpr r -> rows r, r+8)
#pragma unroll
    for (int r = 0; r < 8; ++r) {
      int m0 = 32 * rp + r + 8 * half;
      hmid[m0 * IC + cA * 16 + ln]        = gelu_exact(c1a[r]);
      hmid[(m0 + 16) * IC + cA * 16 + ln] = gelu_exact(c1b[r]);
    }
    __syncthreads();

    // ---- GEMM2: out[MT, nb:nb+NB] += hmid[MT, IC] @ w_out[c0:c0+IC, nb:]
    for (int kb = 0; kb < IC; kb += 4) {
      const int k0 = kb + 2 * half;
      v2f a0 = *(const v2f*)&hmid[(ln)      * IC + k0];
      v2f a1 = *(const v2f*)&hmid[(16 + ln) * IC + k0];
      v2f a2 = *(const v2f*)&hmid[(32 + ln) * IC + k0];
      v2f a3 = *(const v2f*)&hmid[(48 + ln) * IC + k0];
      const float* wr0 = wo + (size_t)(c0 + k0) * kH;
      const float* wr1 = wr0 + kH;
#pragma unroll
      for (int ct = 0; ct < 2; ++ct) {
        const int col = nb + 32 * wv + 16 * ct + ln;
        v2f b; b.x = wr0[col]; b.y = wr1[col];
        acc[0][ct] = wmma_f32_16x16x4(a0, b, acc[0][ct]);
        acc[1][ct] = wmma_f32_16x16x4(a1, b, acc[1][ct]);
        acc[2][ct] = wmma_f32_16x16x4(a2, b, acc[2][ct]);
        acc[3][ct] = wmma_f32_16x16x4(a3, b, acc[3][ct]);
      }
    }
    __syncthreads();
  }

  // scale by routing weight and accumulate (global_atomic_add_f32);
  // padded rows carry scale 0 so they contribute nothing.
#pragma unroll
  for (int rt = 0; rt < 4; ++rt) {
#pragma unroll
    for (int ct = 0; ct < 2; ++ct) {
      const int col = nb + 32 * wv + 16 * ct + ln;
#pragma unroll
      for (int r = 0; r < 8; ++r) {
        int row = 16 * rt + r + 8 * half;
        atomicAdd(&out[(size_t)toks[row] * kH + col], acc[rt][ct][r] * cmbs[row]);
      }
    }
  }
}

extern "C" void kernel_launch(void* const* d_in, const int* in_sizes, int n_in,
                              void* d_out, int out_size, void* d_ws, size_t ws_size,
                              hipStream_t stream) {
  (void)in_sizes; (void)n_in; (void)out_size; (void)ws_size;
  const float* x      = (const float*)d_in[0];  // [B,S,H]
  const float* gate_w = (const float*)d_in[1];  // [H,E]
  const float* w_in   = (const float*)d_in[2];  // [E,H,I]
  const float* w_out  = (const float*)d_in[3];  // [E,I,H]
  const float* bias   = (const float*)d_in[4];  // [H]

  float* out    = (float*)d_out;                // [T,H]
  float* logits = out + (size_t)kT * kH;        // [T,E] (second output)

  // workspace: combine[T,E] | cnts[8] (+pad) | idx[E,T]   (~256.3 KB)
  float* combine = (float*)d_ws;
  int*   cnts    = (int*)d_ws + (size_t)kT * kE;
  int*   idx     = cnts + 64;

  moe_init_out<<<(kT * kH) / 256, 256, 0, stream>>>(out, bias);
  moe_router<<<kT / 256, 256, 0, stream>>>(x, gate_w, logits, combine);
  moe_gather<<<1, 256, 0, stream>>>(combine, cnts, idx);
  dim3 grid(kT / MT, kH / NB, kE);
  moe_expert_ffn<<<grid, 512, 0, stream>>>(x, w_in, w_out, combine, cnts, idx, out);
}